// GSModel_73469710566056
// MI455X (gfx1250) — compile-verified
//
#include <hip/hip_runtime.h>
#include <math.h>
#include <stdint.h>

typedef __attribute__((ext_vector_type(2))) float v2f;
typedef __attribute__((ext_vector_type(8))) float v8f;

#define NG   2048
#define IMW  128
#define IMH  128
#define GSTR 12   // floats per packed gaussian record

// packed record: [ux, uy, ci0, ci1, ci2, alpha, colR, colG, colB, valid, pad, pad]

// ---------------------------------------------------------------------------
// Kernel 1: per-gaussian preprocessing
// ---------------------------------------------------------------------------
__global__ __launch_bounds__(256) void gs_prep(
    const float* __restrict__ pws, const float* __restrict__ shs,
    const float* __restrict__ araw, const float* __restrict__ sraw,
    const float* __restrict__ rraw, const float* __restrict__ Rcw,
    const float* __restrict__ tcw,
    const int* pfx, const int* pfy, const int* pcx, const int* pcy,
    const int* pwid, const int* phgt,
    float* __restrict__ depthOut, float* __restrict__ gsU,
    float* __restrict__ areasOut)
{
    int i = blockIdx.x * 256 + threadIdx.x;
    if (i >= NG) return;

    const float fx = (float)pfx[0], fy = (float)pfy[0];
    const float cx = (float)pcx[0], cy = (float)pcy[0];
    const float width = (float)pwid[0], height = (float)phgt[0];

    const float R00=Rcw[0],R01=Rcw[1],R02=Rcw[2];
    const float R10=Rcw[3],R11=Rcw[4],R12=Rcw[5];
    const float R20=Rcw[6],R21=Rcw[7],R22=Rcw[8];
    const float t0=tcw[0], t1=tcw[1], t2=tcw[2];

    const float px = pws[3*i], py = pws[3*i+1], pz = pws[3*i+2];

    // camera-space point
    const float pcxx = R00*px + R01*py + R02*pz + t0;
    const float pcyy = R10*px + R11*py + R12*pz + t1;
    const float pczz = R20*px + R21*py + R22*pz + t2;

    const float depth = pczz;
    bool valid = depth > 0.2f;
    const float z = valid ? depth : 1.0f;
    const float invz = 1.0f / z;

    const float ux = fx * pcxx * invz + cx;
    const float uy = fy * pcyy * invz + cy;

    // alpha = sigmoid
    const float alpha = 1.0f / (1.0f + expf(-araw[i]));

    // scales
    const float s0 = expf(sraw[3*i]), s1 = expf(sraw[3*i+1]), s2 = expf(sraw[3*i+2]);

    // normalized quaternion -> rotation
    float q0 = rraw[4*i], q1 = rraw[4*i+1], q2 = rraw[4*i+2], q3 = rraw[4*i+3];
    const float qinv = 1.0f / sqrtf(q0*q0 + q1*q1 + q2*q2 + q3*q3);
    const float w = q0*qinv, xq = q1*qinv, yq = q2*qinv, zq = q3*qinv;

    const float G00 = 1.0f - 2.0f*(yq*yq + zq*zq);
    const float G01 = 2.0f*(xq*yq - w*zq);
    const float G02 = 2.0f*(xq*zq + w*yq);
    const float G10 = 2.0f*(xq*yq + w*zq);
    const float G11 = 1.0f - 2.0f*(xq*xq + zq*zq);
    const float G12 = 2.0f*(yq*zq - w*xq);
    const float G20 = 2.0f*(xq*zq - w*yq);
    const float G21 = 2.0f*(yq*zq + w*xq);
    const float G22 = 1.0f - 2.0f*(xq*xq + yq*yq);

    // M = R * diag(s); cov3d = M M^T
    const float M00=G00*s0, M01=G01*s1, M02=G02*s2;
    const float M10=G10*s0, M11=G11*s1, M12=G12*s2;
    const float M20=G20*s0, M21=G21*s1, M22=G22*s2;

    const float V00 = M00*M00 + M01*M01 + M02*M02;
    const float V01 = M00*M10 + M01*M11 + M02*M12;
    const float V02 = M00*M20 + M01*M21 + M02*M22;
    const float V11 = M10*M10 + M11*M11 + M12*M12;
    const float V12 = M10*M20 + M11*M21 + M12*M22;
    const float V22 = M20*M20 + M21*M21 + M22*M22;

    // Jacobian with clipped tangent point
    const float limx = 1.3f * 0.5f * width  / fx;
    const float limy = 1.3f * 0.5f * height / fy;
    const float txc = fminf(fmaxf(pcxx*invz, -limx), limx) * z;
    const float tyc = fminf(fmaxf(pcyy*invz, -limy), limy) * z;
    const float J00 = fx*invz,  J02 = -fx*txc*invz*invz;
    const float J11 = fy*invz,  J12 = -fy*tyc*invz*invz;

    // T = J @ Rcw (2x3)
    const float T00 = J00*R00 + J02*R20;
    const float T01 = J00*R01 + J02*R21;
    const float T02 = J00*R02 + J02*R22;
    const float T10 = J11*R10 + J12*R20;
    const float T11 = J11*R11 + J12*R21;
    const float T12 = J11*R12 + J12*R22;

    // cov2d = T V T^T + 0.3 I
    const float w00 = T00*V00 + T01*V01 + T02*V02;
    const float w01 = T00*V01 + T01*V11 + T02*V12;
    const float w02 = T00*V02 + T01*V12 + T02*V22;
    const float w10 = T10*V00 + T11*V01 + T12*V02;
    const float w11 = T10*V01 + T11*V11 + T12*V12;
    const float w12 = T10*V02 + T11*V12 + T12*V22;
    const float a  = w00*T00 + w01*T01 + w02*T02 + 0.3f;
    const float bq = w00*T10 + w01*T11 + w02*T12;
    const float cc = w10*T10 + w11*T11 + w12*T12 + 0.3f;

    // view direction: twc = -Rcw^T tcw
    const float twcx = -(R00*t0 + R10*t1 + R20*t2);
    const float twcy = -(R01*t0 + R11*t1 + R21*t2);
    const float twcz = -(R02*t0 + R12*t1 + R22*t2);
    float dx = px - twcx, dy = py - twcy, dz = pz - twcz;
    const float dinv = 1.0f / sqrtf(dx*dx + dy*dy + dz*dz);
    const float xn = dx*dinv, yn = dy*dinv, zn = dz*dinv;

    // SH basis (degree 3)
    const float SH_C0 = 0.28209479177387814f;
    const float SH_C1 = 0.4886025119029199f;
    const float C20 = 1.0925484305920792f, C21 = -1.0925484305920792f;
    const float C22 = 0.31539156525252005f, C23 = -1.0925484305920792f;
    const float C24 = 0.5462742152960396f;
    const float C30 = -0.5900435899266435f, C31 = 2.890611442640554f;
    const float C32 = -0.4570457994644658f, C33 = 0.3731763325901154f;
    const float C34 = -0.4570457994644658f, C35 = 1.445305721320277f;
    const float C36 = -0.5900435899266435f;

    const float xx = xn*xn, yy = yn*yn, zz = zn*zn;
    const float xy = xn*yn, yz = yn*zn, xz = xn*zn;
    float basis[16];
    basis[0]  = SH_C0;
    basis[1]  = -SH_C1 * yn;
    basis[2]  =  SH_C1 * zn;
    basis[3]  = -SH_C1 * xn;
    basis[4]  = C20 * xy;
    basis[5]  = C21 * yz;
    basis[6]  = C22 * (2.0f*zz - xx - yy);
    basis[7]  = C23 * xz;
    basis[8]  = C24 * (xx - yy);
    basis[9]  = C30 * yn * (3.0f*xx - yy);
    basis[10] = C31 * xy * zn;
    basis[11] = C32 * yn * (4.0f*zz - xx - yy);
    basis[12] = C33 * zn * (2.0f*zz - 3.0f*xx - 3.0f*yy);
    basis[13] = C34 * xn * (4.0f*zz - xx - yy);
    basis[14] = C35 * zn * (xx - yy);
    basis[15] = C36 * xn * (xx - 3.0f*yy);

    const float* sh = shs + (size_t)i * 48;
    float col[3];
    #pragma unroll
    for (int ch = 0; ch < 3; ++ch) {
        float s = 0.0f;
        #pragma unroll
        for (int k = 0; k < 16; ++k) s += basis[k] * sh[k*3 + ch];
        col[ch] = fmaxf(s + 0.5f, 0.0f);
    }

    // conic + areas
    const float det = a*cc - bq*bq;
    valid = valid && (det > 0.0f);
    const float det_s = valid ? det : 1.0f;
    const float ci0 =  cc / det_s;
    const float ci1 = -bq / det_s;
    const float ci2 =   a / det_s;
    const float mid  = 0.5f * (a + cc);
    const float disc = sqrtf(fmaxf(mid*mid - det, 0.1f));
    const float ar0 = 3.0f * sqrtf(fmaxf(mid + disc, 0.0f));
    const float ar1 = 3.0f * sqrtf(fmaxf(mid - disc, 0.0f));
    areasOut[2*i]   = valid ? ar0 : 0.0f;
    areasOut[2*i+1] = valid ? ar1 : 0.0f;
    depthOut[i] = depth;

    float* o = gsU + (size_t)i * GSTR;
    o[0]=ux; o[1]=uy; o[2]=ci0; o[3]=ci1; o[4]=ci2;
    o[5]=alpha; o[6]=col[0]; o[7]=col[1]; o[8]=col[2];
    o[9]=valid ? 1.0f : 0.0f; o[10]=0.0f; o[11]=0.0f;
}

// ---------------------------------------------------------------------------
// Kernel 2: bitonic argsort by depth (ascending) + gather-reorder records
// ---------------------------------------------------------------------------
__global__ __launch_bounds__(1024) void gs_sort(
    const float* __restrict__ depth, const float* __restrict__ gsU,
    float* __restrict__ gsS)
{
    __shared__ __align__(16) float skey[NG];
    __shared__ __align__(16) int   sidx[NG];
    const int t = threadIdx.x;

    skey[t]        = depth[t];        sidx[t]        = t;
    skey[t + 1024] = depth[t + 1024]; sidx[t + 1024] = t + 1024;
    __syncthreads();

    for (int k = 2; k <= NG; k <<= 1) {
        for (int j = k >> 1; j > 0; j >>= 1) {
            const int i   = ((t & ~(j - 1)) << 1) | (t & (j - 1));
            const int ixj = i | j;
            const bool up = (i & k) == 0;
            const float ki = skey[i], kj = skey[ixj];
            const bool sw = up ? (ki > kj) : (ki < kj);
            if (sw) {
                skey[i] = kj; skey[ixj] = ki;
                const int tmp = sidx[i]; sidx[i] = sidx[ixj]; sidx[ixj] = tmp;
            }
            __syncthreads();
        }
    }

    // gather into sorted order
    for (int e = t; e < NG; e += 1024) {
        const int src = sidx[e];
        const float4* s = (const float4*)(gsU + (size_t)src * GSTR);
        float4* d = (float4*)(gsS + (size_t)e * GSTR);
        d[0] = s[0]; d[1] = s[1]; d[2] = s[2];
    }
}

// ---------------------------------------------------------------------------
// Kernel 3: WMMA rasterizer. Tile = 16x8 pixels; one wave per pixel row.
// power(qx) = k0 + k1*qx + k3*qx^2 evaluated for 16 gaussians x 16 pixels
// with a single V_WMMA_F32_16X16X4_F32 per chunk. Staging of sorted records
// into LDS uses GLOBAL_LOAD_ASYNC_TO_LDS_B128 (ASYNCcnt-tracked DMA).
// ---------------------------------------------------------------------------
__global__ __launch_bounds__(256) void gs_raster(
    const float* __restrict__ gsS, float* __restrict__ img)
{
    __shared__ __align__(16) float lds[256 * GSTR];

    const int lane = threadIdx.x & 31;
    const int wave = threadIdx.x >> 5;
    const int col  = lane & 15;
    const bool low = lane < 16;

    const int x = blockIdx.x * 16 + col;
    const int y = blockIdx.y * 8 + wave;

    const float rx = (float)(blockIdx.x * 16) + 8.0f;  // tile ref x
    const float ry = (float)y + 0.5f;                  // exact row y
    const float qx = ((float)x + 0.5f) - rx;           // in [-7.5, 7.5]

    // B operand (4x16 monomials): VGPR0 = {K0 | K2}, VGPR1 = {K1 | K3}
    v2f bm;
    bm.x = low ? 1.0f : qx * qx;
    bm.y = low ? qx   : 0.0f;

    // per-lane LDS byte offset of this thread's staging slot (addr[31:0] of
    // the flat pointer is the wave-relative LDS offset on gfx1250)
    const unsigned ldsoff =
        (unsigned)(uintptr_t)(const void*)(lds + (size_t)threadIdx.x * GSTR);

    float Trun = 1.0f, Cr = 0.0f, Cg = 0.0f, Cb = 0.0f;

    for (int base = 0; base < NG; base += 256) {
        // async DMA: 48B record per thread, memory -> LDS, no VGPR round-trip
        {
            const float* gsrc = gsS + (size_t)(base + threadIdx.x) * GSTR;
            asm volatile(
                "global_load_async_to_lds_b128 %0, %1, off\n\t"
                "global_load_async_to_lds_b128 %0, %1, off offset:16\n\t"
                "global_load_async_to_lds_b128 %0, %1, off offset:32\n\t"
                :
                : "v"(ldsoff), "v"(gsrc)
                : "memory");
            asm volatile("s_wait_asynccnt 0x0" ::: "memory");
        }
        __syncthreads();

        for (int sub = 0; sub < 256; sub += 16) {
            // A operand: per-lane gaussian = sub + (lane&15)
            const float* ga = lds + (size_t)(sub + col) * GSTR;
            const float ux = ga[0], uy = ga[1];
            const float c0 = ga[2], c1 = ga[3], c2 = ga[4];
            const float ex = ux - rx;
            const float ey = uy - ry;
            const float k0 = -0.5f * (c0*ex*ex + c2*ey*ey) - c1*ex*ey;
            const float k1 = c0*ex + c1*ey;
            const float k3 = -0.5f * c0;

            v2f am;
            am.x = low ? k0 : k3;   // K0 | K2
            am.y = low ? k1 : 0.0f; // K1 | K3

            v8f acc = {};
            acc = __builtin_amdgcn_wmma_f32_16x16x4_f32(
                false, am, false, bm, (short)0, acc, false, false);

            // half-wave local blend: low lanes -> gaussians sub..sub+7,
            // high lanes -> gaussians sub+8..sub+15 (same 16 pixels)
            float lr = 0.0f, lg = 0.0f, lb = 0.0f, lt = 1.0f;
            #pragma unroll
            for (int r = 0; r < 8; ++r) {
                const int gidx = sub + (low ? r : r + 8);
                const float* gb = lds + (size_t)gidx * GSTR;
                const float alpha = gb[5];
                const float cR = gb[6], cG = gb[7], cB = gb[8], vf = gb[9];
                const float p = acc[r];
                float g = alpha * __expf(fminf(p, 0.0f));
                g = fminf(g, 0.99f);
                const bool keep = (p <= 0.0f) && (g >= (1.0f/255.0f)) && (vf > 0.5f);
                g = keep ? g : 0.0f;
                lr += lt * g * cR;
                lg += lt * g * cG;
                lb += lt * g * cB;
                lt *= (1.0f - g);
            }

            // merge halves preserving order: C = CA + tA*CB ; T = tA*tB
            const float oR = __shfl_xor(lr, 16, 32);
            const float oG = __shfl_xor(lg, 16, 32);
            const float oB = __shfl_xor(lb, 16, 32);
            const float oT = __shfl_xor(lt, 16, 32);
            const float CAr = low ? lr : oR, CBr = low ? oR : lr;
            const float CAg = low ? lg : oG, CBg = low ? oG : lg;
            const float CAb = low ? lb : oB, CBb = low ? oB : lb;
            const float tA  = low ? lt : oT, tB  = low ? oT : lt;
            Cr += Trun * (CAr + tA * CBr);
            Cg += Trun * (CAg + tA * CBg);
            Cb += Trun * (CAb + tA * CBb);
            Trun *= tA * tB;
        }
        __syncthreads();
    }

    if (low) {
        const int o = y * IMW + x;
        img[0 * IMH * IMW + o] = Cr;
        img[1 * IMH * IMW + o] = Cg;
        img[2 * IMH * IMW + o] = Cb;
    }
}

// ---------------------------------------------------------------------------
extern "C" void kernel_launch(void* const* d_in, const int* in_sizes, int n_in,
                              void* d_out, int out_size, void* d_ws, size_t ws_size,
                              hipStream_t stream) {
    (void)in_sizes; (void)n_in; (void)out_size; (void)ws_size;
    const float* pws  = (const float*)d_in[0];
    const float* shs  = (const float*)d_in[1];
    const float* araw = (const float*)d_in[2];
    const float* sraw = (const float*)d_in[3];
    const float* rraw = (const float*)d_in[4];
    /* d_in[5] = us (unused) */
    const float* Rcw  = (const float*)d_in[6];
    const float* tcw  = (const float*)d_in[7];
    const int* fx  = (const int*)d_in[8];
    const int* fy  = (const int*)d_in[9];
    const int* cx  = (const int*)d_in[10];
    const int* cy  = (const int*)d_in[11];
    const int* wid = (const int*)d_in[12];
    const int* hgt = (const int*)d_in[13];

    float* out   = (float*)d_out;
    float* img   = out;                      // (3,128,128)
    float* areas = out + 3 * IMH * IMW;      // (2048,2)

    float* ws    = (float*)d_ws;
    float* depth = ws;                       // 2048
    float* gsU   = ws + NG;                  // 2048*12 (unsorted records)
    float* gsS   = gsU + (size_t)NG * GSTR;  // 2048*12 (sorted records)

    gs_prep<<<dim3(NG / 256), dim3(256), 0, stream>>>(
        pws, shs, araw, sraw, rraw, Rcw, tcw,
        fx, fy, cx, cy, wid, hgt, depth, gsU, areas);

    gs_sort<<<dim3(1), dim3(1024), 0, stream>>>(depth, gsU, gsS);

    gs_raster<<<dim3(IMW / 16, IMH / 8), dim3(256), 0, stream>>>(gsS, img);
}